// ApproxCompressor_23175643529363
// MI455X (gfx1250) — compile-verified
//
#include <hip/hip_runtime.h>
#include <math.h>

#define N_ROWS 32
#define CHANS  2
#define LEN    131072
#define TILE   256
#define NPART  (LEN / TILE)        // 512 tiles per row
#define LOG2_NPART 9
#define DELAY_TILES 64             // 16384 / 256
#define EPSV   1e-5f

#define LOG2E  1.44269504088896340736f
#define LN2    0.69314718055994530942f

typedef __attribute__((ext_vector_type(2))) float v2f;
typedef __attribute__((ext_vector_type(8))) float v8f;

// Raw hardware transcendentals (args are range-bounded in this kernel; the
// libm wrappers' ldexp/cndmask range fixups are dead weight here).
__device__ __forceinline__ float fexp2(float x) { return __builtin_amdgcn_exp2f(x); }
__device__ __forceinline__ float flog2(float x) { return __builtin_amdgcn_logf(x); }
__device__ __forceinline__ float fexp(float x)  { return fexp2(x * LOG2E); }
__device__ __forceinline__ float fln(float x)   { return flog2(x) * LN2; }

// alpha^d for d >= 0, 0 for d < 0 — branchless (exponent clamped so the
// unconditional exp2 never overflows; selection is a cheap v_cndmask).
__device__ __forceinline__ float apow_or_zero(int d, float l2a) {
    const int dc = d < 0 ? 0 : d;
    const float v = fexp2((float)dc * l2a);
    return d < 0 ? 0.0f : v;
}

// ---------------------------------------------------------------------------
// Phase 1: one wave per 256-sample tile.
//   energy -> LDS -> 16x16 triangular matmul via 4x V_WMMA_F32_16X16X4_F32
//   -> intra-tile chunk-carry scan (16-lane weighted Hillis-Steele)
//   -> store tile-local y (zero entry carry) + partition end value.
// ---------------------------------------------------------------------------
__global__ __launch_bounds__(256) void k_phase1(const float* __restrict__ x,
                                                const float* __restrict__ z_alpha,
                                                float* __restrict__ ylocal,
                                                float* __restrict__ pend)
{
    __shared__ float eL[8 * TILE];
    const int lane = threadIdx.x & 31;
    const int wave = threadIdx.x >> 5;
    const int tile = blockIdx.x * 8 + wave;
    const int n    = tile >> LOG2_NPART;
    const int p    = tile & (NPART - 1);

    const float* xb0 = x + ((size_t)n * CHANS) * LEN + (size_t)p * TILE;
    const float* xb1 = xb0 + LEN;

    // energy: each lane handles 8 consecutive samples, both channels (float4 loads)
    float4 a0 = ((const float4*)(xb0 + lane * 8))[0];
    float4 a1 = ((const float4*)(xb0 + lane * 8))[1];
    float4 b0 = ((const float4*)(xb1 + lane * 8))[0];
    float4 b1 = ((const float4*)(xb1 + lane * 8))[1];
    float* ew = eL + wave * TILE + lane * 8;
    ew[0] = 0.5f * (a0.x * a0.x + b0.x * b0.x);
    ew[1] = 0.5f * (a0.y * a0.y + b0.y * b0.y);
    ew[2] = 0.5f * (a0.z * a0.z + b0.z * b0.z);
    ew[3] = 0.5f * (a0.w * a0.w + b0.w * b0.w);
    ew[4] = 0.5f * (a1.x * a1.x + b1.x * b1.x);
    ew[5] = 0.5f * (a1.y * a1.y + b1.y * b1.y);
    ew[6] = 0.5f * (a1.z * a1.z + b1.z * b1.z);
    ew[7] = 0.5f * (a1.w * a1.w + b1.w * b1.w);
    __syncthreads();

    const float z     = z_alpha[n];
    const float alpha = 1.0f / (1.0f + fexp(-z));
    const float om    = 1.0f - alpha;
    const float l2a   = flog2(alpha);

    const int hi = lane >> 4;
    const int il = lane & 15;

    // Y(16x16) = M * E, M[i,j] = (i>=j) ? (1-a)*a^(i-j) : 0
    // A layout (16x4 f32): lane<16 -> (i=lane, K=0/1), lane>=16 -> (i=lane-16, K=2/3)
    // B layout (4x16 f32): mirrored; E[j,c] = e[c*16 + j], c = lane&15
    v8f acc = {0.f, 0.f, 0.f, 0.f, 0.f, 0.f, 0.f, 0.f};
    const float* eb = eL + wave * TILE;
#pragma unroll
    for (int k = 0; k < 4; ++k) {
        const int j0 = 4 * k + 2 * hi;
        const int d0 = il - j0;
        v2f A;
        A.x = om * apow_or_zero(d0, l2a);
        A.y = om * apow_or_zero(d0 - 1, l2a);
        float2 bv = *(const float2*)(eb + il * 16 + j0);
        v2f B;
        B.x = bv.x;
        B.y = bv.y;
        acc = __builtin_amdgcn_wmma_f32_16x16x4_f32(false, A, false, B,
                                                    (short)0, acc, false, false);
    }

    // chunk totals T[c] = Y[15,c] live at lane 16+c, VGPR 7
    float t15 = acc[7];
    float Tc  = __shfl(t15, 16 + il, 32);

    // inclusive scan S[c] = T[c] + r*S[c-1], r = alpha^16 (per-step multiplier squares)
    float r  = fexp2(16.0f * l2a);
    float xs = Tc;
    float rp = r;
#pragma unroll
    for (int off = 1; off < 16; off <<= 1) {
        float up = __shfl_up(xs, off, 16);
        xs += (il >= off ? rp : 0.0f) * up;
        rp *= rp;
    }
    float s = __shfl_up(xs, 1, 16);   // exclusive: carry INTO chunk c
    if (il == 0) s = 0.0f;

    // y[i,c] = Y[i,c] + alpha^(i+1) * s[c]   (one exp2, then iterative powers)
    float ap = fexp2((float)(8 * hi + 1) * l2a);
#pragma unroll
    for (int v = 0; v < 8; ++v) {
        acc[v] += ap * s;
        ap *= alpha;
    }

    // store tile-local y: raw per-lane acc order (phase 3 reads the same mapping)
    float* yo = ylocal + (size_t)tile * TILE + lane * 8;
    float4 o0, o1;
    o0.x = acc[0]; o0.y = acc[1]; o0.z = acc[2]; o0.w = acc[3];
    o1.x = acc[4]; o1.y = acc[5]; o1.z = acc[6]; o1.w = acc[7];
    ((float4*)yo)[0] = o0;
    ((float4*)yo)[1] = o1;

    if (lane == 31) pend[tile] = acc[7];   // partition end value (i=15,c=15)
}

// ---------------------------------------------------------------------------
// Phase 2: serial carry scan across partitions, one thread per row (tiny).
//   Carry[p] = carry INTO partition p; S[p] = pend[p] + alpha^256 * S[p-1]
// ---------------------------------------------------------------------------
__global__ void k_phase2(const float* __restrict__ pend,
                         const float* __restrict__ z_alpha,
                         float* __restrict__ carry)
{
    const int n = threadIdx.x;
    const float z     = z_alpha[n];
    const float alpha = 1.0f / (1.0f + fexp(-z));
    const float R     = fexp2(256.0f * flog2(alpha));
    const float* pe = pend  + n * NPART;
    float*       cr = carry + n * NPART;
    float cs = 0.0f;
    for (int p = 0; p < NPART; ++p) {
        cr[p] = cs;
        cs = pe[p] + R * cs;
    }
}

// ---------------------------------------------------------------------------
// Phase 3: finalize envelope (carry + 16384-sample truncation correction),
// soft-knee gain, LDS transpose of gains, coalesced apply to both channels.
// ---------------------------------------------------------------------------
__global__ __launch_bounds__(256) void k_phase3(const float* __restrict__ x,
                                                const float* __restrict__ ylocal,
                                                const float* __restrict__ carry,
                                                const float* __restrict__ lth,
                                                const float* __restrict__ lra,
                                                const float* __restrict__ lkn,
                                                const float* __restrict__ z_alpha,
                                                float* __restrict__ out)
{
    __shared__ float gL[8 * TILE];
    const int lane = threadIdx.x & 31;
    const int wave = threadIdx.x >> 5;
    const int tile = blockIdx.x * 8 + wave;
    const int n    = tile >> LOG2_NPART;
    const int p    = tile & (NPART - 1);
    const int hi   = lane >> 4;
    const int il   = lane & 15;

    const float z     = z_alpha[n];
    const float alpha = 1.0f / (1.0f + fexp(-z));
    const float l2a   = flog2(alpha);
    const float kT    = fexp2(16384.0f * l2a);   // truncation weight (≈0 normally)

    const float C1 = carry[tile];
    float y1[8], y2[8];
    {
        const float4* yp = (const float4*)(ylocal + (size_t)tile * TILE + lane * 8);
        *(float4*)(y1)     = yp[0];
        *(float4*)(y1 + 4) = yp[1];
    }
    float C2 = 0.0f;
    if (p >= DELAY_TILES) {                       // per-wave uniform branch
        const int t2 = tile - DELAY_TILES;
        C2 = carry[t2];
        const float4* yp = (const float4*)(ylocal + (size_t)t2 * TILE + lane * 8);
        *(float4*)(y2)     = yp[0];
        *(float4*)(y2 + 4) = yp[1];
    } else {
#pragma unroll
        for (int v = 0; v < 8; ++v) y2[v] = 0.0f;
    }

    const float lt    = lth[n] - 6.0f;
    const float invR  = 1.0f / (1.0f + fexp(lra[n]));
    const float W     = 0.5f * fexp(lkn[n]);
    const float inv4W = 1.0f / (4.0f * W);
    const float g1    = invR - 1.0f;

    float* gw = gL + wave * TILE;
    // alpha^(pos+1): pos = il*16 + 8*hi + v is consecutive in v
    float ap = fexp2((float)(il * 16 + 8 * hi + 1) * l2a);
#pragma unroll
    for (int v = 0; v < 8; ++v) {
        const int pos = il * 16 + 8 * hi + v;     // sample index within tile
        const float yf1 = y1[v] + ap * C1;
        const float yf2 = y2[v] + ap * C2;
        ap *= alpha;
        const float es  = yf1 - kT * yf2;         // truncated-IIR envelope
        const float le  = fln(es + EPSV);
        const float d   = le - lt;
        const float t   = d + W;
        const float lg_mid = g1 * t * t * inv4W;
        const float lg_abv = g1 * d;
        float lg = (d < -W) ? 0.0f : lg_mid;      // branchless knee select
        lg = (d > W) ? lg_abv : lg;
        gw[pos] = fexp(lg);
    }
    __syncthreads();

    const float* xb0 = x   + ((size_t)n * CHANS) * LEN + (size_t)p * TILE;
    const float* xb1 = xb0 + LEN;
    float*       ob0 = out + ((size_t)n * CHANS) * LEN + (size_t)p * TILE;
    float*       ob1 = ob0 + LEN;

    float4 gA = *(const float4*)(gw + lane * 8);
    float4 gB = *(const float4*)(gw + lane * 8 + 4);
    float4 x00 = ((const float4*)(xb0 + lane * 8))[0];
    float4 x01 = ((const float4*)(xb0 + lane * 8))[1];
    float4 x10 = ((const float4*)(xb1 + lane * 8))[0];
    float4 x11 = ((const float4*)(xb1 + lane * 8))[1];

    float4 r00, r01, r10, r11;
    r00.x = gA.x * x00.x; r00.y = gA.y * x00.y; r00.z = gA.z * x00.z; r00.w = gA.w * x00.w;
    r01.x = gB.x * x01.x; r01.y = gB.y * x01.y; r01.z = gB.z * x01.z; r01.w = gB.w * x01.w;
    r10.x = gA.x * x10.x; r10.y = gA.y * x10.y; r10.z = gA.z * x10.z; r10.w = gA.w * x10.w;
    r11.x = gB.x * x11.x; r11.y = gB.y * x11.y; r11.z = gB.z * x11.z; r11.w = gB.w * x11.w;

    ((float4*)(ob0 + lane * 8))[0] = r00;
    ((float4*)(ob0 + lane * 8))[1] = r01;
    ((float4*)(ob1 + lane * 8))[0] = r10;
    ((float4*)(ob1 + lane * 8))[1] = r11;
}

// ---------------------------------------------------------------------------
extern "C" void kernel_launch(void* const* d_in, const int* in_sizes, int n_in,
                              void* d_out, int out_size, void* d_ws, size_t ws_size,
                              hipStream_t stream)
{
    const float* x   = (const float*)d_in[0];   // [32, 2, 131072]
    const float* lth = (const float*)d_in[1];   // [32, 1]
    const float* lra = (const float*)d_in[2];   // [32, 1]
    const float* lkn = (const float*)d_in[3];   // [32, 1]
    const float* zal = (const float*)d_in[4];   // [32, 1]
    float* out = (float*)d_out;

    // workspace layout: [ylocal: N*L floats][pend: N*NPART][carry: N*NPART]  (~17 MB)
    float* yl   = (float*)d_ws;
    float* pend = yl + (size_t)N_ROWS * LEN;
    float* carr = pend + N_ROWS * NPART;

    const int tiles = N_ROWS * NPART;           // 16384 tiles, 8 waves/block
    k_phase1<<<tiles / 8, 256, 0, stream>>>(x, zal, yl, pend);
    k_phase2<<<1, 32, 0, stream>>>(pend, zal, carr);
    k_phase3<<<tiles / 8, 256, 0, stream>>>(x, yl, carr, lth, lra, lkn, zal, out);
}